// PALACE_prot_net_10900626997619
// MI455X (gfx1250) — compile-verified
//
#include <hip/hip_runtime.h>
#include <hip/hip_bf16.h>

// Problem constants (match reference)
#define VOCAB 33
#define DDIM  256
#define BATCH 64
#define SEQL  4096

typedef __attribute__((ext_vector_type(2))) float v2f;
typedef __attribute__((ext_vector_type(8))) float v8f;

// ---------------------------------------------------------------------------
// Kernel 1: per-batch-row token histogram (LDS atomics) + pooled = counts@emb
// One block per batch row, 256 threads. X read coalesced; emb (33x256, 33KB)
// served from L2/L1.
// ---------------------------------------------------------------------------
__global__ __launch_bounds__(256) void PALACE_pool_hist_kernel(
    const int* __restrict__ X,        // [BATCH, SEQL] token ids 0..32
    const int* __restrict__ vlen,     // [BATCH]
    const float* __restrict__ emb,    // [VOCAB, DDIM]
    float* __restrict__ pooled)       // [BATCH, DDIM]
{
    __shared__ unsigned int cnt[VOCAB];
    __shared__ float fcnt[VOCAB];

    const int b = blockIdx.x;
    const int t = threadIdx.x;

    if (t < VOCAB) cnt[t] = 0u;
    __syncthreads();

    const int n = vlen[b];
    const int* xrow = X + b * SEQL;
#pragma unroll
    for (int i = 0; i < SEQL / 256; ++i) {
        const int l = i * 256 + t;          // coalesced across the wave
        if (l < n) {
            atomicAdd(&cnt[xrow[l]], 1u);   // ds_add_u32
        }
    }
    __syncthreads();

    if (t < VOCAB) fcnt[t] = (float)cnt[t];
    __syncthreads();

    // pooled[b, t] = sum_v counts[v] * emb[v, t]
    float acc = 0.0f;
#pragma unroll
    for (int v = 0; v < VOCAB; ++v) {
        acc = fmaf(fcnt[v], emb[v * DDIM + t], acc);
    }
    pooled[b * DDIM + t] = acc;
}

// ---------------------------------------------------------------------------
// Kernels 2/3: Out = relu(In @ W + bias), In [64,256], W [256,256] row-major
// (W[k][n]), via V_WMMA_F32_16X16X4_F32. One 16x16 output tile per wave,
// K=256 consumed in 64 steps of 4. A tile staged into LDS with the CDNA5
// async copy engine (GLOBAL_LOAD_ASYNC_TO_LDS_B128, tracked by ASYNCcnt):
// all 8 waves of a block share the same M-tile.
//
// Fragment layouts (ISA 7.12.2, f32):
//   A 16x4 : lane m = lane&15 holds k = 2*(lane>>4) and k+1 in v[0],v[1]
//   B 4x16 : lane n = lane&15 holds k = 2*(lane>>4) and k+1 in v[0],v[1]
//   C/D    : VGPR i, lanes 0-15 -> M=i, N=lane ; lanes 16-31 -> M=i+8
// ---------------------------------------------------------------------------
// Row pad: 260 floats -> row stride 1040 B (16B-aligned for B128 LDS writes),
// bank stride 260 % 64 = 4 -> the 16 A-rows land in 16 distinct banks
// (conflict-free for the 16-lane fragment reads), and (m*260 + even_k)*4 keeps
// every v2f fragment read 8B-aligned.
#define A_STRIDE 260

__global__ __launch_bounds__(256) void PALACE_mlp_layer_wmma_kernel(
    const float* __restrict__ In,     // [BATCH, DDIM]
    const float* __restrict__ W,      // [DDIM, DDIM]  (k-major rows)
    const float* __restrict__ bias,   // [DDIM]
    float* __restrict__ Out)          // [BATCH, DDIM]
{
    __shared__ float Atile[16 * A_STRIDE];

    const int t  = threadIdx.x;
    const int mt = blockIdx.x >> 1;        // 0..3  (M-tile, shared per block pair)
    const int m0 = mt * 16;

    // ---- Async stage of the 16x256 fp32 A tile into LDS -------------------
    // 16KB total, 256 threads x 4 iterations x 16B. Both the global source
    // (col multiple of 4 floats) and the LDS destination (row stride 1040B)
    // are 16B-aligned. Writes go straight to LDS (no VGPR round-trip) and are
    // tracked by ASYNCcnt.
#pragma unroll
    for (int i = 0; i < 4; ++i) {
        const int flat = (i * 256 + t) * 4;     // element index, 4 floats/granule
        const int row  = flat >> 8;             // / DDIM
        const int col  = flat & 255;            // % DDIM
        const unsigned lds_dst =
            (unsigned)(uintptr_t)(&Atile[row * A_STRIDE + col]);
        const float* gsrc = &In[(m0 + row) * DDIM + col];
        asm volatile("global_load_async_to_lds_b128 %0, %1, off"
                     :: "v"(lds_dst), "v"(gsrc)
                     : "memory");
    }
    asm volatile("s_wait_asynccnt 0x0" ::: "memory");
    __syncthreads();

    const int wave = t >> 5;               // 0..7
    const int lane = t & 31;
    const int nt   = ((blockIdx.x & 1) << 3) + wave;  // 0..15 (N-tile)
    const int n0   = nt * 16;

    const int mn = lane & 15;              // A row / B col within the tile
    const int kk = (lane >> 4) << 1;       // 0 or 2 (K sub-offset per lane half)

    v8f acc = {};
#pragma unroll 4
    for (int k0 = 0; k0 < DDIM; k0 += 4) {
        // Prefetch the next K-slab of W into cache (global_prefetch_b8).
        if (k0 + 4 < DDIM) {
            __builtin_prefetch(&W[(k0 + 4 + kk) * DDIM + n0 + mn], 0, 3);
        }

        // A fragment: 2 consecutive K values for row (m0+mn), 8B-aligned LDS load.
        v2f a = *(const v2f*)(&Atile[mn * A_STRIDE + k0 + kk]);

        // B fragment: W[k0+kk][n0+mn], W[k0+kk+1][n0+mn]  (coalesced in n).
        v2f bfrag;
        bfrag.x = W[(k0 + kk)     * DDIM + n0 + mn];
        bfrag.y = W[(k0 + kk + 1) * DDIM + n0 + mn];

        // D = A*B + C   (8 args: neg_a, A, neg_b, B, c_mod, C, reuse_a, reuse_b)
        acc = __builtin_amdgcn_wmma_f32_16x16x4_f32(
            false, a, false, bfrag, (short)0, acc, false, false);
    }

    // Epilogue: bias + ReLU, store per C/D layout.
    const float bn    = bias[n0 + mn];
    const int   mbase = m0 + ((lane >> 4) << 3);  // +8 for upper lane half
#pragma unroll
    for (int i = 0; i < 8; ++i) {
        const float v = acc[i] + bn;
        Out[(mbase + i) * DDIM + n0 + mn] = fmaxf(v, 0.0f);
    }
}

// ---------------------------------------------------------------------------
// Launch: pool -> layer1 -> layer2 (stream-ordered).
// ---------------------------------------------------------------------------
extern "C" void kernel_launch(void* const* d_in, const int* in_sizes, int n_in,
                              void* d_out, int out_size, void* d_ws, size_t ws_size,
                              hipStream_t stream) {
    const int*   X    = (const int*)d_in[0];
    const int*   vlen = (const int*)d_in[1];
    const float* emb  = (const float*)d_in[2];
    const float* W1   = (const float*)d_in[3];
    const float* b1   = (const float*)d_in[4];
    const float* W2   = (const float*)d_in[5];
    const float* b2   = (const float*)d_in[6];
    float* out = (float*)d_out;

    float* pooled = (float*)d_ws;                 // [64,256]
    float* h      = pooled + BATCH * DDIM;        // [64,256]

    PALACE_pool_hist_kernel<<<BATCH, 256, 0, stream>>>(X, vlen, emb, pooled);

    // 64 tiles of 16x16 -> 64 waves -> 8 blocks x 8 waves
    PALACE_mlp_layer_wmma_kernel<<<8, 256, 0, stream>>>(pooled, W1, b1, h);
    PALACE_mlp_layer_wmma_kernel<<<8, 256, 0, stream>>>(h, W2, b2, out);
}